// Coords2RMSD_56495999812053
// MI455X (gfx1250) — compile-verified
//
#include <hip/hip_runtime.h>
#include <cstdint>

typedef __attribute__((ext_vector_type(2))) float v2f;
typedef __attribute__((ext_vector_type(8))) float v8f;
typedef __attribute__((ext_vector_type(4))) unsigned int u32x4;
typedef __attribute__((ext_vector_type(4))) int i32x4;
typedef __attribute__((ext_vector_type(8))) int i32x8;

#define NATOMS 2048
#define ROW (3 * NATOMS) /* 6144 floats per row */

__device__ __forceinline__ float wave_sum(float v) {
#pragma unroll
  for (int off = 16; off > 0; off >>= 1) v += __shfl_xor(v, off, 32);
  return v;
}

#if __has_builtin(__builtin_amdgcn_tensor_load_to_lds)
// Build a D# for a 1-row (2D, dim1=1) tile of ROW f32 elements: global -> LDS.
__device__ __forceinline__ void tdm_load_row(uint32_t lds_off, const float* gptr) {
  uint64_t ga = (uint64_t)(uintptr_t)gptr;
  u32x4 g0;
  g0[0] = 1u;                                   // count=1, user mode
  g0[1] = lds_off;                              // lds_addr (bytes)
  g0[2] = (uint32_t)(ga & 0xFFFFFFFFu);         // global_addr[31:0]
  g0[3] = (uint32_t)((ga >> 32) & 0x1FFFFFFu)   // global_addr[56:32]
        | (2u << 30);                           // type=2 ("image")
  i32x8 g1;
  g1[0] = 0x00020000;                           // wg_mask=0, data_size=2 (4B)
  g1[1] = (int)(((uint32_t)ROW & 0xFFFFu) << 16);        // tensor_dim0[15:0] @63:48
  g1[2] = (int)((((uint32_t)ROW >> 16) & 0xFFFFu)        // tensor_dim0[31:16] @79:64
        | (1u << 16));                                   // tensor_dim1=1 @111:80 (lo)
  g1[3] = (int)(((uint32_t)ROW & 0xFFFFu) << 16);        // tile_dim0=ROW @127:112
  g1[4] = 1;                                    // tile_dim1=1, tile_dim2=0
  g1[5] = ROW;                                  // tensor_dim0_stride lo
  g1[6] = 0;                                    // stride hi, dim1_stride lo
  g1[7] = 0;
  i32x4 gz4 = {0, 0, 0, 0};
  i32x8 gz8 = {0, 0, 0, 0, 0, 0, 0, 0};
  __builtin_amdgcn_tensor_load_to_lds(g0, g1, gz4, gz4, gz8, 0);
}
#endif

__global__ __launch_bounds__(256) void kabsch_rmsd_kernel(
    const float* __restrict__ X, const float* __restrict__ Y,
    const int* __restrict__ len, float* __restrict__ out) {
  __shared__ float lx[ROW];
  __shared__ float ly[ROW];
  // acc[0..8]=cov[a][b], [9..11]=Sx, [12..14]=Sy, [15]=S|x|^2, [16]=S|y|^2
  __shared__ float acc[17];

  const int b    = blockIdx.x;
  const int t    = threadIdx.x;
  const int lane = t & 31;
  const int wave = t >> 5;
  const int n    = len[b] + 1;

  if (t < 17) acc[t] = 0.0f;

  const float* xr = X + (size_t)b * ROW;
  const float* yr = Y + (size_t)b * ROW;

#if __has_builtin(__builtin_amdgcn_tensor_load_to_lds)
  if (wave == 0) {
    tdm_load_row((uint32_t)(uintptr_t)lx, xr);
    tdm_load_row((uint32_t)(uintptr_t)ly, yr);
    __builtin_amdgcn_s_wait_tensorcnt(0);
  }
#else
  {
    const float4* x4 = (const float4*)xr;
    const float4* y4 = (const float4*)yr;
    float4* lx4 = (float4*)lx;
    float4* ly4 = (float4*)ly;
    for (int i = t; i < ROW / 4; i += 256) { lx4[i] = x4[i]; ly4[i] = y4[i]; }
  }
#endif
  __syncthreads();

  // ---- Phase 1: centroid sums and sums of squares (masked) ----
  // Thread t owns atoms [8t, 8t+8): 24 contiguous floats = 6 x ds_load_b128.
  float sx0 = 0.f, sx1 = 0.f, sx2 = 0.f, sy0 = 0.f, sy1 = 0.f, sy2 = 0.f;
  float sxx = 0.f, syy = 0.f;
  {
    const float4* lx4 = (const float4*)(lx + t * 24);
    const float4* ly4 = (const float4*)(ly + t * 24);
    float xb[24], yb[24];
#pragma unroll
    for (int i = 0; i < 6; ++i) {
      const float4 vx = lx4[i];
      const float4 vy = ly4[i];
      xb[i * 4 + 0] = vx.x; xb[i * 4 + 1] = vx.y;
      xb[i * 4 + 2] = vx.z; xb[i * 4 + 3] = vx.w;
      yb[i * 4 + 0] = vy.x; yb[i * 4 + 1] = vy.y;
      yb[i * 4 + 2] = vy.z; yb[i * 4 + 3] = vy.w;
    }
#pragma unroll
    for (int i = 0; i < 8; ++i) {
      const int atom = t * 8 + i;
      const float m  = (atom < n) ? 1.0f : 0.0f;
      const float x0 = xb[i * 3 + 0], x1 = xb[i * 3 + 1], x2 = xb[i * 3 + 2];
      const float y0 = yb[i * 3 + 0], y1 = yb[i * 3 + 1], y2 = yb[i * 3 + 2];
      sx0 += m * x0; sx1 += m * x1; sx2 += m * x2;
      sy0 += m * y0; sy1 += m * y1; sy2 += m * y2;
      sxx += m * (x0 * x0 + x1 * x1 + x2 * x2);
      syy += m * (y0 * y0 + y1 * y1 + y2 * y2);
    }
  }
  sx0 = wave_sum(sx0); sx1 = wave_sum(sx1); sx2 = wave_sum(sx2);
  sy0 = wave_sum(sy0); sy1 = wave_sum(sy1); sy2 = wave_sum(sy2);
  sxx = wave_sum(sxx); syy = wave_sum(syy);
  if (lane == 0) {
    atomicAdd(&acc[9], sx0);  atomicAdd(&acc[10], sx1); atomicAdd(&acc[11], sx2);
    atomicAdd(&acc[12], sy0); atomicAdd(&acc[13], sy1); atomicAdd(&acc[14], sy2);
    atomicAdd(&acc[15], sxx); atomicAdd(&acc[16], syy);
  }

  // ---- Phase 2: covariance Sum x_a*y_b via V_WMMA_F32_16X16X4_F32 ----
  // Row/col m = coord*4 + chunk (rows 12..15 zero). Each WMMA covers 16 atoms:
  // atom = base0 + chunk*4 + k. Only chunk-diagonal D entries are read out.
  // Loads are unconditional (addresses clamped in-bounds); masking is done by
  // multiplying with 0/1 so no EXEC-mask branches are generated.
  {
    v8f d = {0.f, 0.f, 0.f, 0.f, 0.f, 0.f, 0.f, 0.f};
    const int base  = wave * 256;
    const int half  = lane >> 4;  // K group: 0 -> K=0,1 ; 1 -> K=2,3
    const int r     = lane & 15;  // A row / B col index
    const int aa    = r >> 2;     // coordinate (3 == zero padding row/col)
    const int ca    = r & 3;      // chunk
    const int cIdx  = (aa < 3) ? aa : 0;  // clamped (safe) coordinate index
    const float rowm = (aa < 3) ? 1.0f : 0.0f;
#pragma unroll 4
    for (int j = 0; j < 16; ++j) {
      const int base0 = base + j * 16;
      v2f av, bv;
#pragma unroll
      for (int v = 0; v < 2; ++v) {
        const int k   = v + 2 * half;
        const int at  = base0 + ca * 4 + k;       // always < NATOMS
        const float m = (at < n) ? rowm : 0.0f;
        av[v] = m * lx[at * 3 + cIdx];
        bv[v] = m * ly[at * 3 + cIdx];
      }
      d = __builtin_amdgcn_wmma_f32_16x16x4_f32(
          false, av, false, bv, (short)0, d, false, false);
    }
    // D[m][ncol]: lane = ncol + 16*(m>=8), reg = m & 7. Diagonal chunk entries:
    if (aa < 3) {
      if (half == 0) {
        atomicAdd(&acc[0 * 3 + aa], d[ca]);      // coord-a 0
        atomicAdd(&acc[1 * 3 + aa], d[4 + ca]);  // coord-a 1
      } else {
        atomicAdd(&acc[2 * 3 + aa], d[ca]);      // coord-a 2
      }
    }
  }
  __syncthreads();

  // ---- Phase 3: 3x3 closed-form SVD trace + RMSD (one lane) ----
  if (t == 0) {
    const double nn = (double)n;
    double C[3][3];
    for (int i = 0; i < 3; ++i)
      for (int j = 0; j < 3; ++j) C[i][j] = (double)acc[i * 3 + j];
    double cx[3], cy[3];
    for (int i = 0; i < 3; ++i) {
      cx[i] = (double)acc[9 + i] / nn;
      cy[i] = (double)acc[12 + i] / nn;
    }
    double ssq = (double)acc[15] + (double)acc[16];
    for (int i = 0; i < 3; ++i) ssq -= nn * (cx[i] * cx[i] + cy[i] * cy[i]);
    for (int i = 0; i < 3; ++i)
      for (int j = 0; j < 3; ++j) C[i][j] -= nn * cx[i] * cy[j];

    const double detC =
        C[0][0] * (C[1][1] * C[2][2] - C[1][2] * C[2][1]) -
        C[0][1] * (C[1][0] * C[2][2] - C[1][2] * C[2][0]) +
        C[0][2] * (C[1][0] * C[2][1] - C[1][1] * C[2][0]);

    double A[3][3];  // A = C^T C (symmetric PSD)
    for (int i = 0; i < 3; ++i)
      for (int j = 0; j < 3; ++j) {
        double s = 0.0;
        for (int k = 0; k < 3; ++k) s += C[k][i] * C[k][j];
        A[i][j] = s;
      }
    const double q  = (A[0][0] + A[1][1] + A[2][2]) / 3.0;
    const double p1 = A[0][1] * A[0][1] + A[0][2] * A[0][2] + A[1][2] * A[1][2];
    const double p2 = (A[0][0] - q) * (A[0][0] - q) +
                      (A[1][1] - q) * (A[1][1] - q) +
                      (A[2][2] - q) * (A[2][2] - q) + 2.0 * p1;
    double e0, e1, e2;
    if (p2 <= 1e-30) {
      e0 = e1 = e2 = q;
    } else {
      const double p   = sqrt(p2 / 6.0);
      const double inv = 1.0 / p;
      const double B00 = (A[0][0] - q) * inv, B11 = (A[1][1] - q) * inv,
                   B22 = (A[2][2] - q) * inv;
      const double B01 = A[0][1] * inv, B02 = A[0][2] * inv, B12 = A[1][2] * inv;
      double detB = B00 * (B11 * B22 - B12 * B12) -
                    B01 * (B01 * B22 - B12 * B02) +
                    B02 * (B01 * B12 - B11 * B02);
      double rr = 0.5 * detB;
      rr = fmin(1.0, fmax(-1.0, rr));
      const double phi = acos(rr) / 3.0;
      e0 = q + 2.0 * p * cos(phi);
      e2 = q + 2.0 * p * cos(phi + 2.0943951023931953);  // +2pi/3
      e1 = 3.0 * q - e0 - e2;
    }
    const double s0 = sqrt(fmax(e0, 0.0));
    const double s1 = sqrt(fmax(e1, 0.0));
    const double s2 = sqrt(fmax(e2, 0.0));
    const double dsig = (detC >= 0.0) ? 1.0 : -1.0;
    const double tr  = s0 + s1 + dsig * s2;
    const double msd = fmax(ssq - 2.0 * tr, 0.0) / nn;
    out[b] = (float)sqrt(msd + 1e-12);
  }
}

extern "C" void kernel_launch(void* const* d_in, const int* in_sizes, int n_in,
                              void* d_out, int out_size, void* d_ws, size_t ws_size,
                              hipStream_t stream) {
  const float* X = (const float*)d_in[0];
  const float* Y = (const float*)d_in[1];
  const int* L   = (const int*)d_in[2];
  float* out     = (float*)d_out;
  const int B    = in_sizes[2];  // 4096 batch elements
  kabsch_rmsd_kernel<<<B, 256, 0, stream>>>(X, Y, L, out);
}